// MultiHeadAttention_82497731822190
// MI455X (gfx1250) — compile-verified
//
#include <hip/hip_runtime.h>

typedef float v2f __attribute__((ext_vector_type(2)));
typedef float v8f __attribute__((ext_vector_type(8)));
typedef int   i32x4 __attribute__((vector_size(16)));   // matches builtin's param type

#define D_MODEL 512
#define NUM_HEADS 8
#define DEPTH 64
#define BATCH 16
#define SEQ 1024

// ---- CDNA5 async global->LDS support (guarded; sync fallback keeps compile green)
#if defined(__has_builtin)
#if __has_builtin(__builtin_amdgcn_global_load_async_to_lds_b128) && \
    __has_builtin(__builtin_amdgcn_global_load_async_to_lds_b32)  && \
    __has_builtin(__builtin_amdgcn_s_wait_asynccnt)
#define USE_ASYNC_LDS 1
#endif
#endif
#ifndef USE_ASYNC_LDS
#define USE_ASYNC_LDS 0
#endif

#if USE_ASYNC_LDS
__device__ __forceinline__ void async_b128(const float* g, float* l) {
    __builtin_amdgcn_global_load_async_to_lds_b128(
        (__attribute__((address_space(1))) i32x4*)g,
        (__attribute__((address_space(3))) i32x4*)l, 0, 0);
}
__device__ __forceinline__ void async_b32(const float* g, float* l) {
    __builtin_amdgcn_global_load_async_to_lds_b32(
        (__attribute__((address_space(1))) int*)g,
        (__attribute__((address_space(3))) int*)l, 0, 0);
}
#endif

// ---------------------------------------------------------------------------
// fp32 WMMA GEMM with double-buffered async LDS pipeline:
//   C[z] = alpha * A[z] @ B[z]
//   A: [M,K] row-major; B: [K,N] row-major, or TRANSB: source [N,K] transposed
//   into LDS. B tile kept in fragment-native packed layout Bpk[BK/4][BN][4] so
//   each lane's (k0,k0+1) pair is one ds_load_b64.
//   Wave32: each wave computes 32x32 (2x2 of 16x16 f32 WMMA accumulators).
// ---------------------------------------------------------------------------
template <int BM, int BN, int BK, int WROWS, int WCOLS, bool TRANSB>
__global__ __launch_bounds__(32 * WROWS * WCOLS)
void gemm_wmma_f32(const float* __restrict__ A, int lda, long long sA0, long long sA1,
                   const float* __restrict__ B, int ldb, long long sB0, long long sB1,
                   float* __restrict__ C, int ldc, long long sC0, long long sC1,
                   int zdiv, int K, float alpha)
{
    constexpr int NTHREADS = 32 * WROWS * WCOLS;
    constexpr int BK4 = BK / 4;
#if USE_ASYNC_LDS
    constexpr int A_OPS = (BM * BK4 + NTHREADS - 1) / NTHREADS;                 // b128
    constexpr int B_OPS = TRANSB ? (BN * BK4 + NTHREADS - 1) / NTHREADS         // b128
                                 : (BK * BN + NTHREADS - 1) / NTHREADS;         // b32
    constexpr int FILL_OPS = A_OPS + B_OPS;
#endif

    // +4 float pad -> row stride 80B: keeps every 16B async chunk 16B-aligned
    __shared__ float As[2][BM][BK + 4];
    __shared__ float Bpk[2][BK4][BN][4];

    const int tid  = threadIdx.x;
    const int lane = tid & 31;
    const int w    = tid >> 5;
    const int wrow = w / WCOLS;
    const int wcol = w % WCOLS;
    const int half = lane >> 4;   // selects K-pair (A/B) / upper-M-half (C/D)
    const int ln   = lane & 15;

    const int zo = blockIdx.z / zdiv;
    const int zi = blockIdx.z % zdiv;
    const float* Ab = A + (long long)zo * sA0 + (long long)zi * sA1;
    const float* Bb = B + (long long)zo * sB0 + (long long)zi * sB1;
    float*       Cb = C + (long long)zo * sC0 + (long long)zi * sC1;

    const int bm0 = blockIdx.y * BM;
    const int bn0 = blockIdx.x * BN;

    auto fill = [&](int s, int k0) {
#if USE_ASYNC_LDS
        for (int idx = tid; idx < BM * BK4; idx += NTHREADS) {
            int r = idx / BK4, c4 = (idx % BK4) * 4;
            async_b128(&Ab[(long long)(bm0 + r) * lda + k0 + c4], &As[s][r][c4]);
        }
        if constexpr (TRANSB) {
            for (int idx = tid; idx < BN * BK4; idx += NTHREADS) {
                int n = idx / BK4, c4 = (idx % BK4) * 4;
                async_b128(&Bb[(long long)(bn0 + n) * ldb + k0 + c4],
                           &Bpk[s][c4 >> 2][n][0]);
            }
        } else {
            for (int idx = tid; idx < BK * BN; idx += NTHREADS) {
                int r = idx / BN, c = idx % BN;
                async_b32(&Bb[(long long)(k0 + r) * ldb + bn0 + c],
                          &Bpk[s][r >> 2][c][r & 3]);
            }
        }
#else
        for (int idx = tid; idx < BM * BK; idx += NTHREADS) {
            int r = idx / BK, c = idx % BK;
            As[s][r][c] = Ab[(long long)(bm0 + r) * lda + (k0 + c)];
        }
        if constexpr (TRANSB) {
            for (int idx = tid; idx < BK * BN; idx += NTHREADS) {
                int n = idx / BK, c = idx % BK;
                Bpk[s][c >> 2][n][c & 3] = Bb[(long long)(bn0 + n) * ldb + (k0 + c)];
            }
        } else {
            for (int idx = tid; idx < BK * BN; idx += NTHREADS) {
                int r = idx / BN, c = idx % BN;
                Bpk[s][r >> 2][c][r & 3] = Bb[(long long)(k0 + r) * ldb + (bn0 + c)];
            }
        }
#endif
    };

    v8f acc[2][2] = {};

    const int nstage = K / BK;
    fill(0, 0);

    for (int st = 0; st < nstage; ++st) {
        const int sb = st & 1;
        // kick off next stage into the other buffer while this one computes
        if (st + 1 < nstage) fill(sb ^ 1, (st + 1) * BK);
#if USE_ASYNC_LDS
        if (st + 1 < nstage) __builtin_amdgcn_s_wait_asynccnt(FILL_OPS);
        else                 __builtin_amdgcn_s_wait_asynccnt(0);
#endif
        __syncthreads();

        #pragma unroll
        for (int kk = 0; kk < BK; kk += 4) {
            v2f afr[2], bfr[2];
            #pragma unroll
            for (int t = 0; t < 2; ++t) {
                int m = wrow * 32 + 16 * t + ln;
                afr[t] = *(const v2f*)&As[sb][m][kk + 2 * half];   // ds_load_b64
            }
            #pragma unroll
            for (int u = 0; u < 2; ++u) {
                int n = wcol * 32 + 16 * u + ln;
                bfr[u] = *(const v2f*)&Bpk[sb][kk >> 2][n][2 * half]; // ds_load_b64
            }
            #pragma unroll
            for (int t = 0; t < 2; ++t)
                #pragma unroll
                for (int u = 0; u < 2; ++u)
                    acc[t][u] = __builtin_amdgcn_wmma_f32_16x16x4_f32(
                        false, afr[t], false, bfr[u], (short)0, acc[t][u],
                        false, false);
        }
        __syncthreads();   // protect buffer sb from being refilled at st+2
    }

    // C/D 16x16 f32 layout: VGPR v holds M = v + 8*half, N = ln
    #pragma unroll
    for (int t = 0; t < 2; ++t)
        for (int u = 0; u < 2; ++u)
            #pragma unroll
            for (int v = 0; v < 8; ++v) {
                int m = bm0 + wrow * 32 + 16 * t + v + 8 * half;
                int n = bn0 + wcol * 32 + 16 * u + ln;
                Cb[(long long)m * ldc + n] = acc[t][u][v] * alpha;
            }
}

// ---------------------------------------------------------------------------
// In-place masked softmax over rows of length SEQ (one block per row).
// ---------------------------------------------------------------------------
__global__ __launch_bounds__(256)
void softmax_mask_kernel(float* __restrict__ attn, const float* __restrict__ mask)
{
    __shared__ float red[256];
    const int tid = threadIdx.x;
    const long long row = blockIdx.x;
    float* p = attn + row * (long long)SEQ;
    const int b = (int)(row / ((long long)NUM_HEADS * SEQ));
    const float* mrow = mask + (long long)b * SEQ;

    float v[4];
    float mx = -3.402823466e38f;
    #pragma unroll
    for (int i = 0; i < 4; ++i) {
        int c = tid + i * 256;
        float x = p[c] + (1.0f - mrow[c]) * (-1.0e9f);
        v[i] = x;
        mx = fmaxf(mx, x);
    }
    red[tid] = mx;
    __syncthreads();
    for (int s = 128; s > 0; s >>= 1) {
        if (tid < s) red[tid] = fmaxf(red[tid], red[tid + s]);
        __syncthreads();
    }
    mx = red[0];
    __syncthreads();

    float sum = 0.f;
    #pragma unroll
    for (int i = 0; i < 4; ++i) { v[i] = __expf(v[i] - mx); sum += v[i]; }
    red[tid] = sum;
    __syncthreads();
    for (int s = 128; s > 0; s >>= 1) {
        if (tid < s) red[tid] += red[tid + s];
        __syncthreads();
    }
    const float inv = 1.0f / red[0];
    #pragma unroll
    for (int i = 0; i < 4; ++i) p[tid + i * 256] = v[i] * inv;
}

// ---------------------------------------------------------------------------
extern "C" void kernel_launch(void* const* d_in, const int* in_sizes, int n_in,
                              void* d_out, int out_size, void* d_ws, size_t ws_size,
                              hipStream_t stream)
{
    (void)in_sizes; (void)n_in; (void)out_size; (void)ws_size;

    const float* v_in = (const float*)d_in[0];
    const float* k_in = (const float*)d_in[1];
    const float* q_in = (const float*)d_in[2];
    const float* mask = (const float*)d_in[3];
    const float* Wq   = (const float*)d_in[4];
    const float* Wk   = (const float*)d_in[5];
    const float* Wv   = (const float*)d_in[6];
    const float* Wo   = (const float*)d_in[7];

    const long long BSD = (long long)BATCH * SEQ * D_MODEL;   // 8,388,608
    float* out  = (float*)d_out;       // [B,S,D]
    float* attn = out + BSD;           // [B,H,S,S]

    float* qh = (float*)d_ws;          // [B*S, D]  (later reused as ctx)
    float* kh = qh + BSD;
    float* vh = kh + BSD;
    float* ctx = qh;                   // safe: qh dead after logits GEMM

    const int M = BATCH * SEQ;         // 16384
    const dim3 blk(256);
    const float rsqrt_dk = 0.125f;     // 1/sqrt(64)

    // 1-3) Q/K/V projections: [16384,512] x [512,512]
    {
        dim3 g(D_MODEL / 128, M / 64, 1);
        gemm_wmma_f32<64, 128, 16, 2, 4, false><<<g, blk, 0, stream>>>(
            q_in, D_MODEL, 0, 0, Wq, D_MODEL, 0, 0, qh, D_MODEL, 0, 0,
            1, D_MODEL, 1.0f);
        gemm_wmma_f32<64, 128, 16, 2, 4, false><<<g, blk, 0, stream>>>(
            k_in, D_MODEL, 0, 0, Wk, D_MODEL, 0, 0, kh, D_MODEL, 0, 0,
            1, D_MODEL, 1.0f);
        gemm_wmma_f32<64, 128, 16, 2, 4, false><<<g, blk, 0, stream>>>(
            v_in, D_MODEL, 0, 0, Wv, D_MODEL, 0, 0, vh, D_MODEL, 0, 0,
            1, D_MODEL, 1.0f);
    }

    // 4) logits[b,h] = (Qh @ Kh^T) / sqrt(dk) ; z = b*H + h
    {
        dim3 g(SEQ / 128, SEQ / 64, BATCH * NUM_HEADS);
        const long long sQ0 = (long long)SEQ * D_MODEL, sQ1 = DEPTH;
        const long long sC0 = (long long)NUM_HEADS * SEQ * SEQ, sC1 = (long long)SEQ * SEQ;
        gemm_wmma_f32<64, 128, 16, 2, 4, true><<<g, blk, 0, stream>>>(
            qh, D_MODEL, sQ0, sQ1,
            kh, D_MODEL, sQ0, sQ1,
            attn, SEQ, sC0, sC1,
            NUM_HEADS, DEPTH, rsqrt_dk);
    }

    // 5) masked softmax over key dim, in place
    {
        dim3 g((unsigned)((long long)BATCH * NUM_HEADS * SEQ));
        softmax_mask_kernel<<<g, blk, 0, stream>>>(attn, mask);
    }

    // 6) ctx[b,h] = attn[b,h] @ Vh[b,h]   -> ctx laid out [b,s, h*64+d]
    {
        dim3 g(DEPTH / 64, SEQ / 128, BATCH * NUM_HEADS);
        const long long sA0 = (long long)NUM_HEADS * SEQ * SEQ, sA1 = (long long)SEQ * SEQ;
        const long long sV0 = (long long)SEQ * D_MODEL, sV1 = DEPTH;
        gemm_wmma_f32<128, 64, 16, 4, 2, false><<<g, blk, 0, stream>>>(
            attn, SEQ, sA0, sA1,
            vh, D_MODEL, sV0, sV1,
            ctx, D_MODEL, sV0, sV1,
            NUM_HEADS, SEQ, 1.0f);
    }

    // 7) out = ctx @ Wo : [16384,512] x [512,512]
    {
        dim3 g(D_MODEL / 128, M / 64, 1);
        gemm_wmma_f32<64, 128, 16, 2, 4, false><<<g, blk, 0, stream>>>(
            ctx, D_MODEL, 0, 0, Wo, D_MODEL, 0, 0, out, D_MODEL, 0, 0,
            1, D_MODEL, 1.0f);
    }
}